// ModulePopHistory_2027224564325
// MI455X (gfx1250) — compile-verified
//
#include <hip/hip_runtime.h>
#include <math.h>

typedef __attribute__((ext_vector_type(2))) float v2f;
typedef __attribute__((ext_vector_type(8))) float v8f;

#define TSTEPS 128
#define BLOCK  128            // 4 waves of 32
#define WAVES  4
#define ROWS_PER_BLOCK 64     // 16 rows per wave
#define XPAD   132            // row stride in floats (bank-conflict-free pad)
#define ALPHA_F 0.2f
#define CDEC_F  0.8f

__global__ __launch_bounds__(BLOCK) void pop_ema_wmma_kernel(
    const float* __restrict__ x,     // [N, 128]
    const int*   __restrict__ timev, // [N]
    float*       __restrict__ out,   // [N]
    int n)
{
    __shared__ float xt[ROWS_PER_BLOCK * XPAD];   // 33,792 B : staged rows
    __shared__ float Wt[TSTEPS * 16];             //  8,192 B : weights W[t][n]
    __shared__ float Et[WAVES * 16 * 16];         //  4,096 B : block-end EMAs

    const int tid  = threadIdx.x;
    const int lane = tid & 31;
    const int wave = tid >> 5;
    const int hi   = lane >> 4;   // K-half selector for wave32 WMMA layouts
    const int rm   = lane & 15;
    const int r0   = blockIdx.x * ROWS_PER_BLOCK;

    // ---- Build weight table: W[t][n] = coefficient of x_t in e[16n+15] ----
    // e[m] = c^m * x0 + a * sum_{t=1..m} c^(m-t) * x_t
    for (int i = tid; i < TSTEPS * 16; i += BLOCK) {
        int t = i >> 4, nn = i & 15;
        float w = 0.0f;
        if (nn < 8) {
            int m = 16 * nn + 15;
            if (t <= m)
                w = (t == 0) ? powf(CDEC_F, (float)m)
                             : ALPHA_F * powf(CDEC_F, (float)(m - t));
        }
        Wt[i] = w;
    }

    // ---- Stage 64x128 f32 tile into LDS, coalesced float4 ----
    for (int it = 0; it < 16; ++it) {
        int lin  = it * BLOCK + tid;      // float4 index within tile (2048 total)
        int row  = lin >> 5;              // 32 float4 per 128-float row
        int c4   = lin & 31;
        int grow = r0 + row;
        if (grow >= n) grow = n - 1;      // safety clamp (N%64==0 so never taken)
        float4 v = ((const float4*)(x + (size_t)grow * TSTEPS))[c4];
        *(float4*)&xt[row * XPAD + c4 * 4] = v;
    }
    __syncthreads();

    // ---- WMMA: E[16 rows x 8 block-ends] = X[16x128] @ W[128x16] in f32 ----
    const float* xw = &xt[wave * 16 * XPAD];
    v8f acc = {0.f, 0.f, 0.f, 0.f, 0.f, 0.f, 0.f, 0.f};
#pragma unroll
    for (int ks = 0; ks < TSTEPS / 4; ++ks) {
        int kb = ks * 4;
        // A (16x4 f32): lanes 0-15 hold row rm, K={kb,kb+1}; lanes 16-31 K={kb+2,kb+3}
        v2f a = *(const v2f*)&xw[rm * XPAD + kb + 2 * hi];
        // B (4x16 f32): lane holds col rm; VGPR0/1 = K-pair of its half
        int tb = kb + 2 * hi;
        v2f b;
        b.x = Wt[tb * 16 + rm];
        b.y = Wt[(tb + 1) * 16 + rm];
        acc = __builtin_amdgcn_wmma_f32_16x16x4_f32(
            /*neg_a=*/false, a, /*neg_b=*/false, b,
            /*c_mod=*/(short)0, acc, /*reuse_a=*/false, /*reuse_b=*/false);
    }

    // ---- Stage D tile to LDS: D vgpr i = row (i + 8*hi), col rm ----
    float* Ew = &Et[wave * 256];
#pragma unroll
    for (int vg = 0; vg < 8; ++vg) {
        Ew[(vg + 8 * hi) * 16 + rm] = acc[vg];
    }
    __syncthreads();

    // ---- Per-row finish: continue exact recurrence from nearest block-end ----
    if (lane < 16) {
        int row = wave * 16 + lane;
        int gr  = r0 + row;
        int grc = (gr < n) ? gr : (n - 1);
        int idx = timev[grc] - 1;
        if (idx < 0) idx = 0;
        if (idx > TSTEPS - 2) idx = TSTEPS - 2;

        const float* xr = &xt[row * XPAD];
        float e;
        int tstart;
        if (idx >= 15) {
            int B = (idx - 15) >> 4;      // largest block-end m=16B+15 <= idx
            e = Ew[lane * 16 + B];
            tstart = 16 * B + 16;
        } else {
            e = xr[0];                    // e[0] = x[0]
            tstart = 1;
        }
        for (int t = tstart; t <= idx; ++t)
            e = CDEC_F * e + ALPHA_F * xr[t];   // exact reference recurrence tail

        if (gr < n)
            out[gr] = 1.0f / (1.0f + expf(-e)); // sigmoid
    }
}

extern "C" void kernel_launch(void* const* d_in, const int* in_sizes, int n_in,
                              void* d_out, int out_size, void* d_ws, size_t ws_size,
                              hipStream_t stream) {
    const float* x     = (const float*)d_in[0];   // pop_history [N,128] f32
    const int*   timev = (const int*)d_in[1];     // time [N] i32
    (void)d_ws; (void)ws_size; (void)n_in;
    float* out = (float*)d_out;                   // sigmoid(e) [N] f32
    int n = in_sizes[1];                          // N items
    int blocks = (n + ROWS_PER_BLOCK - 1) / ROWS_PER_BLOCK;  // 15625 for N=1e6
    pop_ema_wmma_kernel<<<blocks, BLOCK, 0, stream>>>(x, timev, out, n);
}